// BaseMOE_84086869721240
// MI455X (gfx1250) — compile-verified
//
#include <hip/hip_runtime.h>
#include <math.h>

typedef __attribute__((ext_vector_type(2))) float v2f;
typedef __attribute__((ext_vector_type(8))) float v8f;

#define EXPERTS 16
#define BATCH   256
#define DIM     1024
#define KPRED   4097
#define VOCAB   50257
#define VP1     50258
#define LN_EPS  1e-6f

// ---------------------------------------------------------------------------
// Fused GEMM(16xN rowblock) + bias + ELU + LayerNorm(ddof=1) layer.
// Block = 256 threads = 8 wave32s. Block computes rows [16*bid, 16*bid+16).
// Wave w owns NT consecutive 16x16 N-tiles; K-loop step 4 via
// V_WMMA_F32_16X16X4_F32 (fp32 matrix pipe, no precision loss).
// A-frag layout (32b A 16x4): lane l, vgpr v -> M = l%16, K = v + 2*(l/16).
// B/C/D: row striped across lanes within a VGPR (C/D: M = r + 8*(l/16)).
// ---------------------------------------------------------------------------
template <int K, int N, int NT>
__global__ __launch_bounds__(256) void mlp_layer(
    const float* __restrict__ X,     // (M, K) row-major
    const float* __restrict__ W,     // (K, N) row-major
    const float* __restrict__ bias,  // (N)
    const float* __restrict__ gam,   // (N)
    const float* __restrict__ bet,   // (N)
    float* __restrict__ Y)           // (M, N)
{
    __shared__ float tile[16 * N];

    const int lane = threadIdx.x & 31;
    const int wave = threadIdx.x >> 5;
    const int half = lane >> 4;   // 0: lanes 0-15, 1: lanes 16-31
    const int lrow = lane & 15;
    const int row0 = blockIdx.x * 16;

    v8f acc[NT];
#pragma unroll
    for (int t = 0; t < NT; ++t) acc[t] = (v8f){0.f,0.f,0.f,0.f,0.f,0.f,0.f,0.f};

    const float* __restrict__ arow = X + (size_t)(row0 + lrow) * K;

    for (int k = 0; k < K; k += 4) {
        // A fragment: two consecutive fp32 (K = k+2*half, k+2*half+1), 8B load
        v2f a = *(const v2f*)(arow + k + 2 * half);
#pragma unroll
        for (int t = 0; t < NT; ++t) {
            const int col = (wave * NT + t) * 16 + lrow;
            v2f b;
            b.x = W[(size_t)(k + 2 * half) * N + col];
            b.y = W[(size_t)(k + 2 * half + 1) * N + col];
            acc[t] = __builtin_amdgcn_wmma_f32_16x16x4_f32(
                false, a, false, b, (short)0, acc[t], false, false);
        }
    }

    // bias + ELU -> LDS (rows of the 16xN block)
#pragma unroll
    for (int t = 0; t < NT; ++t) {
        const int col = (wave * NT + t) * 16 + lrow;
        const float bc = bias[col];
#pragma unroll
        for (int r = 0; r < 8; ++r) {
            float v = acc[t][r] + bc;
            v = (v > 0.f) ? v : expm1f(v);
            tile[(r + 8 * half) * N + col] = v;
        }
    }
    __syncthreads();

    // LayerNorm: wave w -> rows 2w (lanes 0-15) and 2w+1 (lanes 16-31).
    {
        const int row = 2 * wave + half;
        const float* __restrict__ trow = tile + row * N;

        float s = 0.f;
#pragma unroll 4
        for (int j = lrow; j < N; j += 16) s += trow[j];
#pragma unroll
        for (int m = 8; m >= 1; m >>= 1) s += __shfl_xor(s, m, 16);
        const float mu = s * (1.f / (float)N);

        float sq = 0.f;
#pragma unroll 4
        for (int j = lrow; j < N; j += 16) {
            const float d = trow[j] - mu;
            sq += d * d;
        }
#pragma unroll
        for (int m = 8; m >= 1; m >>= 1) sq += __shfl_xor(sq, m, 16);
        const float sd  = sqrtf(sq * (1.f / (float)(N - 1)));   // ddof=1
        const float inv = 1.f / (sd + LN_EPS);                  // (sd + eps)

        float* __restrict__ yrow = Y + (size_t)(row0 + row) * N;
#pragma unroll 4
        for (int j = lrow; j < N; j += 16) {
            yrow[j] = gam[j] * ((trow[j] - mu) * inv) + bet[j];
        }
    }
}

// ---------------------------------------------------------------------------
// logits = H3 @ Wout + bout  (128 -> 1), then softmax over the B axis per
// expert. One block per expert, one thread per batch element.
// ---------------------------------------------------------------------------
__global__ __launch_bounds__(256) void logits_softmax(
    const float* __restrict__ H,     // (E*B, 128)
    const float* __restrict__ Wout,  // (128)
    const float* __restrict__ bout,  // (1)
    float* __restrict__ routing)     // (E*B)
{
    const int e = blockIdx.x;
    const int b = threadIdx.x;
    const float* __restrict__ h = H + (size_t)(e * BATCH + b) * 128;

    float s = 0.f;
#pragma unroll 8
    for (int j = 0; j < 128; ++j) s += h[j] * Wout[j];
    s += bout[0];

    __shared__ float red[256];
    red[b] = s;
    __syncthreads();
#pragma unroll
    for (int off = 128; off >= 1; off >>= 1) {
        if (b < off) red[b] = fmaxf(red[b], red[b + off]);
        __syncthreads();
    }
    const float mx = red[0];
    __syncthreads();

    const float ex = expf(s - mx);
    red[b] = ex;
    __syncthreads();
#pragma unroll
    for (int off = 128; off >= 1; off >>= 1) {
        if (b < off) red[b] += red[b + off];
        __syncthreads();
    }
    routing[e * BATCH + b] = ex / red[0];
}

// ---------------------------------------------------------------------------
// Initialize out (B, V+1, 2): channel0 = 0, channel1 = v (v<V) else -1.
// One float2 (b64) store per (b,v).
// ---------------------------------------------------------------------------
__global__ __launch_bounds__(256) void init_out(float* __restrict__ out)
{
    const size_t i = (size_t)blockIdx.x * blockDim.x + threadIdx.x;
    const size_t total = (size_t)BATCH * VP1;
    if (i >= total) return;
    const int v = (int)(i % VP1);
    float2 val;
    val.x = 0.f;
    val.y = (v < VOCAB) ? (float)v : -1.f;
    ((float2*)out)[i] = val;
}

// ---------------------------------------------------------------------------
// Scatter-add: out[b, idx[e,b,k], 0] += routing[e,b] * probs[e,b,k], k<4096.
// Coalesced float2 stream of `prediction`; fp32 global atomics.
// ---------------------------------------------------------------------------
__global__ __launch_bounds__(256) void scatter_add(
    const float* __restrict__ pred,     // (E, B, KPRED, 2)
    const float* __restrict__ routing,  // (E*B)
    float* __restrict__ out)            // (B, V+1, 2)
{
    const unsigned i = blockIdx.x * blockDim.x + threadIdx.x;  // < 16*256*4096
    const int kk = i & 4095;
    const unsigned rest = i >> 12;
    const int b = rest & 255;
    const int e = rest >> 8;

    const size_t pi = ((size_t)(e * BATCH + b) * KPRED + kk) * 2;
    const float2 pv = *(const float2*)(pred + pi);
    const float val = routing[e * BATCH + b] * pv.x;
    const int tgt = (int)pv.y;
    atomicAdd(out + ((size_t)b * VP1 + tgt) * 2, val);
}

// ---------------------------------------------------------------------------
extern "C" void kernel_launch(void* const* d_in, const int* in_sizes, int n_in,
                              void* d_out, int out_size, void* d_ws, size_t ws_size,
                              hipStream_t stream)
{
    const float* emb  = (const float*)d_in[0];   // (16,256,1024)
    const float* pred = (const float*)d_in[1];   // (16,256,4097,2)
    const float* W1   = (const float*)d_in[2];
    const float* b1   = (const float*)d_in[3];
    const float* g1   = (const float*)d_in[4];
    const float* be1  = (const float*)d_in[5];
    const float* W2   = (const float*)d_in[6];
    const float* b2   = (const float*)d_in[7];
    const float* g2   = (const float*)d_in[8];
    const float* be2  = (const float*)d_in[9];
    const float* W3   = (const float*)d_in[10];
    const float* b3   = (const float*)d_in[11];
    const float* g3   = (const float*)d_in[12];
    const float* be3  = (const float*)d_in[13];
    const float* Wout = (const float*)d_in[14];
    const float* bout = (const float*)d_in[15];

    float* out = (float*)d_out;

    // workspace layout (floats)
    float* H1      = (float*)d_ws;                 // 4096*512
    float* H2      = H1 + (size_t)4096 * 512;      // 4096*256
    float* H3      = H2 + (size_t)4096 * 256;      // 4096*128
    float* routing = H3 + (size_t)4096 * 128;      // 4096

    const int rowBlocks = (EXPERTS * BATCH) / 16;  // 256

    mlp_layer<1024, 512, 4><<<rowBlocks, 256, 0, stream>>>(emb, W1, b1, g1, be1, H1);
    mlp_layer< 512, 256, 2><<<rowBlocks, 256, 0, stream>>>(H1,  W2, b2, g2, be2, H2);
    mlp_layer< 256, 128, 1><<<rowBlocks, 256, 0, stream>>>(H2,  W3, b3, g3, be3, H3);

    logits_softmax<<<EXPERTS, 256, 0, stream>>>(H3, Wout, bout, routing);

    const size_t initElems = (size_t)BATCH * VP1;                // b64 stores
    init_out<<<(unsigned)((initElems + 255) / 256), 256, 0, stream>>>(out);

    const unsigned scatterThreads = EXPERTS * BATCH * 4096u;     // 16.78M
    scatter_add<<<scatterThreads / 256, 256, 0, stream>>>(pred, routing, out);
}